// ChamferDistance_11218454577160
// MI455X (gfx1250) — compile-verified
//
#include <hip/hip_runtime.h>
#include <hip/hip_bf16.h>
#include <math.h>

typedef __attribute__((ext_vector_type(2))) float v2f;
typedef __attribute__((ext_vector_type(8))) float v8f;

#define WAVES_PER_BLOCK 8   // 256 threads, 8 wave32 waves
#define COLSPLIT 8          // column-range splits per row group

// Thread i < BN: pack pred point i as {x,y,z,|p|^2}; i in [BN,2BN): pack gt.
// Also initializes all 2*BN min slots to +inf bits.
__global__ void chamfer_prep_kernel(const float* __restrict__ pred,
                                    const float* __restrict__ gt,
                                    unsigned int* __restrict__ mins,
                                    float4* __restrict__ pred4,
                                    float4* __restrict__ gt4,
                                    int BN) {
    int i = blockIdx.x * blockDim.x + threadIdx.x;
    if (i >= 2 * BN) return;
    mins[i] = 0x7F800000u;  // +inf
    const int isPred = (i < BN);
    const float* src = isPred ? (pred + (size_t)i * 3)
                              : (gt + (size_t)(i - BN) * 3);
    float x0 = src[0], x1 = src[1], x2 = src[2];
    float4 v = make_float4(x0, x1, x2, x0 * x0 + x1 * x1 + x2 * x2);
    if (isPred) pred4[i] = v;
    else        gt4[i - BN] = v;
}

// For each row x in X (per batch), min over y in Y of ||x-y||^2 via
// V_WMMA_F32_16X16X4_F32 with the norm terms folded into the K=4 slot:
//   A row  = [-2x0, -2x1, -2x2, 1]
//   B col  = [ y0,   y1,   y2,  |y|^2]
//   c[i][j] = -2*x.y + |y|^2 = d2[i][j] - |x_i|^2
// Inner loop keeps only min(rmin, c); +g2 and clamp hoisted out (both commute
// with min). rowmin_sq holds float bits as uint (values >= 0 after clamp).
__global__ void chamfer_rowmin_kernel(const float4* __restrict__ X4,
                                      const float4* __restrict__ Y4,
                                      unsigned int* __restrict__ rowmin_sq,
                                      int N, int B) {
    const int lane = threadIdx.x & 31;
    const int hi   = lane >> 4;      // 0: lanes 0-15 (K=0,1), 1: lanes 16-31 (K=2,3)
    const int sub  = lane & 15;
    const int wave = blockIdx.x * WAVES_PER_BLOCK + (threadIdx.x >> 5);

    const int ntile  = N >> 4;
    const int split  = wave % COLSPLIT;
    const int rowgrp = (wave / COLSPLIT) % ntile;
    const int b      = wave / (COLSPLIT * ntile);
    if (b >= B) return;

    const int row_base = rowgrp << 4;
    const float4* __restrict__ Xb = X4 + (size_t)b * N;
    const float4* __restrict__ Yb = Y4 + (size_t)b * N;

    // --- A tile (16x4 f32): lanes 0-15: (-2x0,-2x1); lanes 16-31: (-2x2, 1) ---
    const float* ap = (const float*)(Xb + (row_base + sub)) + (hi << 1);
    v2f a;
    a.x = -2.0f * ap[0];
    a.y = hi ? 1.0f : (-2.0f * ap[1]);

    // --- per-lane |x_row|^2 for the 8 C rows this lane covers (row_base+r+8*hi)
    float g2r[8];
#pragma unroll
    for (int r = 0; r < 8; ++r)
        g2r[r] = ((const float*)(Xb + (row_base + r + (hi << 3))))[3];

    float rmin[8];
#pragma unroll
    for (int r = 0; r < 8; ++r) rmin[r] = __builtin_inff();

    const int tiles_per_split = ntile / COLSPLIT;
    const int jt0 = split * tiles_per_split;

    // Per-lane column pointer: lanes 0-15 read {y0,y1}, lanes 16-31 {y2,|y|^2}
    const float* yp = (const float*)(Yb + ((jt0 << 4) + sub)) + (hi << 1);

    for (int t = 0; t < tiles_per_split; ++t) {
        v2f bm;
        bm.x = yp[0];
        bm.y = yp[1];
        yp += 64;                                   // 16 points * 4 floats

        v8f c = {};
        c = __builtin_amdgcn_wmma_f32_16x16x4_f32(
                /*neg_a=*/false, a, /*neg_b=*/false, bm,
                /*c_mod=*/(short)0, c, /*reuse_a=*/false, /*reuse_b=*/false);

#pragma unroll
        for (int r = 0; r < 8; ++r)
            rmin[r] = fminf(rmin[r], c[r]);
    }

    // Min-reduce across the 16 lanes of each half (xor masks 1..8 stay in-half)
#pragma unroll
    for (int r = 0; r < 8; ++r) {
        float v = rmin[r];
#pragma unroll
        for (int m = 8; m >= 1; m >>= 1)
            v = fminf(v, __shfl_xor(v, m, 32));
        rmin[r] = v;
    }

    if (sub == 0) {
#pragma unroll
        for (int r = 0; r < 8; ++r) {
            int row = row_base + r + (hi << 3);
            float d2 = fmaxf(rmin[r] + g2r[r], 0.0f);   // restore g2, clamp
            atomicMin(&rowmin_sq[(size_t)b * N + row], __float_as_uint(d2));
        }
    }
}

// Single-block deterministic reduction:
// out = (1/(B*N)) * sum over all 2*B*N entries of sqrt(min_sq)
__global__ void chamfer_finalize_kernel(const unsigned int* __restrict__ dsq,
                                        float* __restrict__ out,
                                        int total, float scale) {
    __shared__ float smem[1024];
    float acc = 0.0f;
    for (int i = threadIdx.x; i < total; i += blockDim.x)
        acc += sqrtf(__uint_as_float(dsq[i]));
    smem[threadIdx.x] = acc;
    __syncthreads();
    for (int off = blockDim.x >> 1; off > 0; off >>= 1) {
        if ((int)threadIdx.x < off) smem[threadIdx.x] += smem[threadIdx.x + off];
        __syncthreads();
    }
    if (threadIdx.x == 0) out[0] = smem[0] * scale;
}

extern "C" void kernel_launch(void* const* d_in, const int* in_sizes, int n_in,
                              void* d_out, int out_size, void* d_ws, size_t ws_size,
                              hipStream_t stream) {
    const float* pred = (const float*)d_in[0];  // [B, N, 3]
    const float* gt   = (const float*)d_in[1];  // [B, N, 3]
    float* out = (float*)d_out;

    const int B = 2;
    const int N = in_sizes[0] / (3 * B);        // 8192
    const int BN = B * N;

    // Workspace layout (640 KB total):
    //   [0, 2*BN) uint   : d1sq mins (first BN) then d2sq mins (next BN)
    //   then BN float4   : packed pred {x,y,z,|p|^2}
    //   then BN float4   : packed gt
    unsigned int* mins = (unsigned int*)d_ws;
    unsigned int* d1sq = mins;
    unsigned int* d2sq = mins + BN;
    float4* pred4 = (float4*)((char*)d_ws + (size_t)2 * BN * sizeof(unsigned int));
    float4* gt4   = pred4 + BN;

    chamfer_prep_kernel<<<(2 * BN + 255) / 256, 256, 0, stream>>>(
        pred, gt, mins, pred4, gt4, BN);

    const int ntile = N >> 4;
    const int nwaves = B * ntile * COLSPLIT;
    const int nblocks = nwaves / WAVES_PER_BLOCK;

    // dist1: rows = gt, cols = pred
    chamfer_rowmin_kernel<<<nblocks, WAVES_PER_BLOCK * 32, 0, stream>>>(gt4, pred4, d1sq, N, B);
    // dist2: rows = pred, cols = gt
    chamfer_rowmin_kernel<<<nblocks, WAVES_PER_BLOCK * 32, 0, stream>>>(pred4, gt4, d2sq, N, B);

    // d1sq and d2sq are contiguous: reduce all 2*B*N entries at once.
    chamfer_finalize_kernel<<<1, 1024, 0, stream>>>(mins, out, 2 * BN,
                                                    1.0f / (float)(B * N));
}